// ClassCapsuleLayer_23424751633175
// MI455X (gfx1250) — compile-verified
//
#include <hip/hip_runtime.h>

typedef __attribute__((ext_vector_type(2))) float v2f;
typedef __attribute__((ext_vector_type(8))) float v8f;

#define EPSF 1e-7f

constexpr int B_ = 32, H_ = 8, W_ = 8, I_ = 32, O_ = 64;
constexpr int S_  = B_ * H_ * W_;   // 2048 samples
constexpr int N_  = H_ * W_ * I_;   // 2048 routing rows per batch
constexpr int OP_ = O_ * 16;        // 1024 (o,p) stats per batch

// ---------------------------------------------------------------------------
// Kernel 1: votes via V_WMMA_F32_16X16X4_F32.
// Per input-capsule i: GEMM  A(16-row tiles over (s,a), K=4=b)  x  B(4 x (o,c))
// One wave = one 16-row M-tile of one i. All 16 B tiles (4 KB) hoisted into
// registers; coordinate-add constants + store bases precomputed per row so the
// inner loop is branch-free: 16 x (wmma + 8 add + 8 store with imm offsets).
// ---------------------------------------------------------------------------
__global__ __launch_bounds__(256) void votes_wmma_kernel(
    const float* __restrict__ pose,   // (S, I, 4, 4)
    const float* __restrict__ wmat,   // (I, O, 4, 4)
    float* __restrict__ votes)        // (B, N, O, 16)
{
    const int wave = blockIdx.x * (256 / 32) + (threadIdx.x >> 5);
    const int lane = threadIdx.x & 31;

    const int i  = wave >> 9;        // 512 M-tiles per i
    const int mt = wave & 511;
    const int s_base = mt * 4;       // 4 samples per 16-row tile

    const int lm = lane & 15;
    const int hi = lane >> 4;        // half-wave select
    const int k0 = hi * 2;           // lanes 0-15 hold K=0,1; lanes 16-31 K=2,3

    // A tile: row m = (s_local, a); elements (m,k0),(m,k0+1)
    const int sA = s_base + (lm >> 2);
    const int aA = lm & 3;
    const float* ap = pose + ((size_t)(sA * I_ + i) * 16 + aA * 4 + k0);
    v2f Av;
    Av[0] = ap[0];
    Av[1] = ap[1];

    // B/N-tile lane mapping: n = lm -> (o_local = n/4, c = n%4)
    const int o_off = lm >> 2;
    const int cB    = lm & 3;

    // Hoist ALL B tiles for this i (16 tiles x 2 floats per lane = 32 VGPRs)
    v2f Bv[16];
    #pragma unroll
    for (int nt = 0; nt < 16; ++nt) {
        const int o = nt * 4 + o_off;
        const float* bp = wmat + ((size_t)(i * O_ + o) * 16 + k0 * 4 + cB);
        Bv[nt][0] = bp[0];   // w[i][o][k0  ][c]
        Bv[nt][1] = bp[4];   // w[i][o][k0+1][c]
    }

    // Precompute per-row store offsets and coordinate-add constants
    // (invariant across nt: only o changes with nt, adding nt*64 floats).
    int   offs[8];
    float addv[8];
    #pragma unroll
    for (int r = 0; r < 8; ++r) {
        const int m  = r + hi * 8;
        const int s  = s_base + (m >> 2);
        const int a  = m & 3;
        const int e  = a * 4 + cB;        // element index in 16
        const int bb = s >> 6;            // s / (H*W)
        const int hw = s & 63;
        const int hh = hw >> 3;
        const int ww = hw & 7;
        const int nrow = hw * I_ + i;
        offs[r] = (bb * N_ + nrow) * OP_ + o_off * 16 + e;
        addv[r] = (e == 0) ? (hh + 0.5f) * 0.125f
                : (e == 1) ? (ww + 0.5f) * 0.125f
                           : 0.0f;
    }

    #pragma unroll
    for (int nt = 0; nt < 16; ++nt) {
        v8f Cv = {};
        v8f Dv = __builtin_amdgcn_wmma_f32_16x16x4_f32(
            /*neg_a=*/false, Av, /*neg_b=*/false, Bv[nt],
            /*c_mod=*/(short)0, Cv, /*reuse_a=*/false, /*reuse_b=*/false);
        #pragma unroll
        for (int r = 0; r < 8; ++r) {
            votes[(size_t)offs[r] + nt * 64] = Dv[r] + addv[r];
        }
    }
}

// ---------------------------------------------------------------------------
// Kernel 2: per-batch EM stats. One block per batch, 1024 threads = (o,p).
// Each thread privately reduces S0/S1/S2 over N -> mean/stdv in a single
// votes pass. Per-o sums via width-16 shfl_xor (wave32), cost mean/std via LDS.
// ---------------------------------------------------------------------------
__global__ __launch_bounds__(1024) void stats_kernel(
    const float* __restrict__ votes,   // (B, N, OP_)
    const float* __restrict__ act,     // (B, N)
    const float* __restrict__ rr,      // (B, N, O)
    const float* __restrict__ beta_v,  // (O)
    const float* __restrict__ beta_a,  // (O)
    float* __restrict__ meanb,         // (B, OP_)
    float* __restrict__ i2vb,          // (B, OP_)  = 1/(2 var)
    float* __restrict__ logsumb,       // (B, O)    = sum_p log(stdv+EPS)
    float* __restrict__ oactb,         // (B, O)
    float inv_temp, int first_iter, int write_out,
    float* __restrict__ out_poses,     // (B, O, 16)
    float* __restrict__ out_act)       // (B, O)
{
    const int b = blockIdx.x;
    const int t = threadIdx.x;
    const int o = t >> 4;
    const int p = t & 15;

    const float* vb = votes + (size_t)b * N_ * OP_ + t;
    const float* ab = act + (size_t)b * N_;
    const float* rb = rr + (size_t)b * N_ * O_ + o;

    float s0 = 0.f, s1 = 0.f, s2 = 0.f;
    #pragma unroll 2
    for (int n = 0; n < N_; ++n) {
        const float rv = first_iter ? (1.0f / O_) : rb[(size_t)n * O_];
        const float rp = rv * ab[n];
        const float v  = vb[(size_t)n * OP_];
        s0 += rp;
        s1 += rp * v;
        s2 += rp * v * v;
    }

    const float mean  = s1 / s0;
    const float var   = fmaxf(s2 / s0 - mean * mean, 0.0f);
    const float stdv  = sqrtf(var);
    const float lstd  = logf(stdv + EPSF);
    float cost_h = (beta_v[o] + lstd) * s0;
    float lsum   = lstd;
    #pragma unroll
    for (int off = 8; off >= 1; off >>= 1) {
        cost_h += __shfl_xor(cost_h, off, 16);
        lsum   += __shfl_xor(lsum,   off, 16);
    }

    __shared__ float sc[O_];
    if (p == 0) sc[o] = cost_h;
    __syncthreads();

    float cm = 0.f;
    #pragma unroll 8
    for (int k = 0; k < O_; ++k) cm += sc[k];
    cm *= (1.0f / O_);
    float cs = 0.f;
    #pragma unroll 8
    for (int k = 0; k < O_; ++k) { const float d = sc[k] - cm; cs += d * d; }
    cs = sqrtf(cs * (1.0f / O_));

    const float oa =
        1.0f / (1.0f + expf(-inv_temp * (beta_a[o] + (cm - cost_h) / (cs + EPSF))));

    meanb[(size_t)b * OP_ + t] = mean;
    i2vb[(size_t)b * OP_ + t]  = 0.5f / var;
    if (p == 0) {
        logsumb[b * O_ + o] = lsum;
        oactb[b * O_ + o]   = oa;
    }
    if (write_out) {
        out_poses[((size_t)b * O_ + o) * 16 + p] = mean;   // poses = o_mean
        if (p == 0) out_act[b * O_ + o] = oa;              // activations
    }
}

// ---------------------------------------------------------------------------
// Kernel 3: rr update (softmax over O). One thread per sample row n.
// Per-batch stats staged in LDS; zz cached in padded LDS (stride 65 => no
// bank conflicts on the 64-bank LDS).
// ---------------------------------------------------------------------------
__global__ __launch_bounds__(128) void rr_kernel(
    const float* __restrict__ votes,
    const float* __restrict__ meanb,
    const float* __restrict__ i2vb,
    const float* __restrict__ logsumb,
    const float* __restrict__ oactb,
    float* __restrict__ rr)
{
    __shared__ float sm_mean[OP_];
    __shared__ float sm_i2v[OP_];
    __shared__ float sm_ls[O_];
    __shared__ float sm_la[O_];
    __shared__ float sm_zz[128][O_ + 1];

    const int bpb = N_ / 128;                 // 16 blocks per batch
    const int b   = blockIdx.x / bpb;
    const int n   = (blockIdx.x % bpb) * 128 + threadIdx.x;
    const int tid = threadIdx.x;

    for (int k = tid; k < OP_; k += 128) {
        sm_mean[k] = meanb[(size_t)b * OP_ + k];
        sm_i2v[k]  = i2vb[(size_t)b * OP_ + k];
    }
    if (tid < O_) {
        sm_ls[tid] = logsumb[b * O_ + tid];
        sm_la[tid] = logf(oactb[b * O_ + tid] + EPSF);
    }
    __syncthreads();

    const float4* vrow =
        (const float4*)(votes + ((size_t)(b * N_) + n) * OP_);

    float mx = -3.4e38f;
    for (int oo = 0; oo < O_; ++oo) {
        float acc = 0.f;
        #pragma unroll
        for (int q = 0; q < 4; ++q) {
            const float4 v = vrow[oo * 4 + q];
            const int base = oo * 16 + q * 4;
            float d;
            d = v.x - sm_mean[base + 0]; acc += d * d * sm_i2v[base + 0];
            d = v.y - sm_mean[base + 1]; acc += d * d * sm_i2v[base + 1];
            d = v.z - sm_mean[base + 2]; acc += d * d * sm_i2v[base + 2];
            d = v.w - sm_mean[base + 3]; acc += d * d * sm_i2v[base + 3];
        }
        const float zz = sm_la[oo] - acc - sm_ls[oo];   // log(o_act+EPS) + o_p
        sm_zz[tid][oo] = zz;
        mx = fmaxf(mx, zz);
    }

    float ssum = 0.f;
    for (int oo = 0; oo < O_; ++oo) {
        const float e = expf(sm_zz[tid][oo] - mx);
        sm_zz[tid][oo] = e;
        ssum += e;
    }
    const float inv = 1.0f / ssum;

    float* rout = rr + ((size_t)(b * N_) + n) * O_;
    for (int oo = 0; oo < O_; ++oo) rout[oo] = sm_zz[tid][oo] * inv;
}

// ---------------------------------------------------------------------------
extern "C" void kernel_launch(void* const* d_in, const int* in_sizes, int n_in,
                              void* d_out, int out_size, void* d_ws, size_t ws_size,
                              hipStream_t stream) {
    const float* pose = (const float*)d_in[0];   // (B,H,W,I,4,4)
    const float* act  = (const float*)d_in[1];   // (B,H,W,I)
    const float* wmat = (const float*)d_in[2];   // (I,O,4,4)
    const float* bv   = (const float*)d_in[3];   // (1,O)
    const float* ba   = (const float*)d_in[4];   // (1,O)
    float* out = (float*)d_out;

    float* ws      = (float*)d_ws;
    float* votes   = ws;                                    // B*N*OP  = 67,108,864
    float* rr      = votes + (size_t)B_ * N_ * OP_;         // B*N*O   =  4,194,304
    float* meanb   = rr + (size_t)B_ * N_ * O_;             // B*OP
    float* i2vb    = meanb + (size_t)B_ * OP_;              // B*OP
    float* logsumb = i2vb + (size_t)B_ * OP_;               // B*O
    float* oactb   = logsumb + (size_t)B_ * O_;             // B*O

    float* out_poses = out;                 // (B,O,16)
    float* out_act   = out + B_ * O_ * 16;  // (B,O)

    // Stage 1: votes via WMMA (16384 waves)
    votes_wmma_kernel<<<2048, 256, 0, stream>>>(pose, wmat, votes);

    // Stage 2: 3 EM-routing iterations; inv_temp = 1 + it
    for (int it = 0; it < 3; ++it) {
        const float inv_temp = 1.0f + (float)it;
        stats_kernel<<<B_, 1024, 0, stream>>>(
            votes, act, rr, bv, ba, meanb, i2vb, logsumb, oactb,
            inv_temp, it == 0 ? 1 : 0, it == 2 ? 1 : 0, out_poses, out_act);
        if (it < 2) {
            rr_kernel<<<B_ * (N_ / 128), 128, 0, stream>>>(
                votes, meanb, i2vb, logsumb, oactb, rr);
        }
    }
}